// NonLocalBlock_28329604284751
// MI455X (gfx1250) — compile-verified
//
#include <hip/hip_runtime.h>
#include <math.h>

// ---------------------------------------------------------------------------
// NonLocalBlock for MI455X (gfx1250, wave32).
// B=4, C=128, H=W=64, N=4096. fp32 throughout (tau=0.01 softmax is
// precision-sensitive); matrix math via V_WMMA_F32_16X16X4_F32.
// ---------------------------------------------------------------------------

typedef __attribute__((ext_vector_type(2))) float v2f;
typedef __attribute__((ext_vector_type(8))) float v8f;

constexpr int NB   = 4;     // batch
constexpr int NC   = 128;   // channels
constexpr int NN   = 4096;  // H*W
constexpr float TAU_INV = 100.0f;   // 1/0.01
constexpr float FEPS    = 1e-12f;

// ---------------------------------------------------------------------------
// Kernel 1: 1x1-conv projection  Y[b][o][n] = bias[o] + sum_c W[o][c]*X[b][c][n]
// transpose_out=1 stores Y[b][n][o] (used for the query path so corr A-tiles
// are row-contiguous).  Tiny FLOP count -> plain VALU.
// block (32,8): lanes sweep n (coalesced), y sweeps o (wave-uniform W row).
// ---------------------------------------------------------------------------
__global__ __launch_bounds__(256) void proj_kernel(
    const float* __restrict__ X, const float* __restrict__ W,
    const float* __restrict__ bias, float* __restrict__ Y, int transpose_out) {
  int n = blockIdx.x * 32 + threadIdx.x;
  int o = blockIdx.y * 8 + threadIdx.y;
  int b = blockIdx.z;
  const float* x = X + (size_t)b * NC * NN;
  const float* wrow = W + o * NC;
  float acc = bias[o];
  for (int c = 0; c < NC; ++c)
    acc = fmaf(wrow[c], x[(size_t)c * NN + n], acc);
  if (transpose_out)
    Y[((size_t)b * NN + n) * NC + o] = acc;
  else
    Y[((size_t)b * NC + o) * NN + n] = acc;
}

// ---------------------------------------------------------------------------
// Kernel 2: per-(b,c) spatial means of pq ([b][n][c]) and pk ([b][c][n]).
// ---------------------------------------------------------------------------
__global__ __launch_bounds__(256) void mean_kernel(
    const float* __restrict__ pq, const float* __restrict__ pk,
    float* __restrict__ mq, float* __restrict__ mk) {
  int c = blockIdx.x, b = blockIdx.y, t = threadIdx.x;
  __shared__ float red[256];

  float s = 0.f;
  const float* q = pq + (size_t)b * NN * NC + c;
  for (int n = t; n < NN; n += 256) s += q[(size_t)n * NC];
  red[t] = s; __syncthreads();
  for (int off = 128; off > 0; off >>= 1) {
    if (t < off) red[t] += red[t + off];
    __syncthreads();
  }
  if (t == 0) mq[b * NC + c] = red[0] * (1.0f / NN);
  __syncthreads();

  s = 0.f;
  const float* k = pk + ((size_t)b * NC + c) * NN;
  for (int n = t; n < NN; n += 256) s += k[n];
  red[t] = s; __syncthreads();
  for (int off = 128; off > 0; off >>= 1) {
    if (t < off) red[t] += red[t + off];
    __syncthreads();
  }
  if (t == 0) mk[b * NC + c] = red[0] * (1.0f / NN);
}

// ---------------------------------------------------------------------------
// Kernel 3: center + L2-normalize query rows in place, pq [b][n][c].
// One wave per pixel; lane holds 4 contiguous channels; shfl_xor reduce (w32).
// ---------------------------------------------------------------------------
__global__ __launch_bounds__(256) void norm_q_kernel(
    float* __restrict__ pq, const float* __restrict__ mq) {
  int w = threadIdx.x >> 5, lane = threadIdx.x & 31;
  int q = blockIdx.x * 8 + w, b = blockIdx.y;
  float* row = pq + ((size_t)b * NN + q) * NC;
  const float* m = mq + b * NC;
  float v[4]; float ss = 0.f;
#pragma unroll
  for (int i = 0; i < 4; ++i) {
    int c = lane * 4 + i;
    v[i] = row[c] - m[c];
    ss += v[i] * v[i];
  }
#pragma unroll
  for (int off = 16; off > 0; off >>= 1) ss += __shfl_xor(ss, off, 32);
  float inv = 1.0f / fmaxf(sqrtf(ss), FEPS);
#pragma unroll
  for (int i = 0; i < 4; ++i) row[lane * 4 + i] = v[i] * inv;
}

// ---------------------------------------------------------------------------
// Kernel 4: center + L2-normalize key columns in place, pk [b][c][n].
// Thread = pixel k; channel loop strides by NN (coalesced across lanes).
// ---------------------------------------------------------------------------
__global__ __launch_bounds__(256) void norm_k_kernel(
    float* __restrict__ pk, const float* __restrict__ mk) {
  int k = blockIdx.x * 256 + threadIdx.x;
  int b = blockIdx.y;
  float* base = pk + (size_t)b * NC * NN + k;
  const float* m = mk + b * NC;
  float ss = 0.f;
  for (int c = 0; c < NC; ++c) {
    float v = base[(size_t)c * NN] - m[c];
    ss += v * v;
  }
  float inv = 1.0f / fmaxf(sqrtf(ss), FEPS);
  for (int c = 0; c < NC; ++c)
    base[(size_t)c * NN] = (base[(size_t)c * NN] - m[c]) * inv;
}

// ---------------------------------------------------------------------------
// Kernel 5: corr GEMM (WMMA fp32) + fused softmax + conf_map.
// Block = 256 threads (8 waves) owns a 16-row q-tile x full 4096 k columns.
// Phase A: each wave computes 32 n-tiles of 16x16 via v_wmma_f32_16x16x4_f32
//          (K=128 in 32 steps), writes raw cosine correlations to d_out att,
//          tracks per-row maxima.
// Phase B/C: after a device fence + barrier, each wave owns 2 rows: exp
//          ((x-max)*100), row-sum, normalize.  The 256KB block stays in L2.
// ---------------------------------------------------------------------------
__global__ __launch_bounds__(256) void attn_kernel(
    const float* __restrict__ pqn,   // [b][q][c] normalized
    const float* __restrict__ pkn,   // [b][c][k] normalized
    float* __restrict__ att,         // [b][q][k] output attention
    float* __restrict__ conf) {      // [b][q]
  int b = blockIdx.y;
  int q0 = blockIdx.x * 16;
  int tid = threadIdx.x, wave = tid >> 5, lane = tid & 31;
  int half = lane >> 4, l15 = lane & 15;

  __shared__ float aq[16][130];          // A rows (padded)
  __shared__ float rowmax_p[16][128];    // per-(row, wave*16+halflane) maxima
  __shared__ float rowred[16];           // final row maxima

  // Stage the 16 normalized query rows (A operand).
  for (int i = tid; i < 16 * NC; i += 256) {
    int m = i >> 7, c = i & 127;
    aq[m][c] = pqn[((size_t)b * NN + q0 + m) * NC + c];
  }
  __syncthreads();

  const float* pkb = pkn + (size_t)b * NC * NN;
  float* attRow = att + ((size_t)b * NN + q0) * NN;

  float lmax8[8];
#pragma unroll
  for (int r = 0; r < 8; ++r) lmax8[r] = -3.0e38f;

  // --- Phase A: WMMA correlation, write logits, track row maxima ---
  for (int nt = wave; nt < NN / 16; nt += 8) {
    int k0 = nt * 16;
    v8f acc = {};
    for (int c0 = 0; c0 < NC; c0 += 4) {
      int ck = c0 + half * 2;
      v2f a, bf;
      a.x = aq[l15][ck];
      a.y = aq[l15][ck + 1];
      bf.x = pkb[(size_t)ck * NN + k0 + l15];
      bf.y = pkb[(size_t)(ck + 1) * NN + k0 + l15];
      acc = __builtin_amdgcn_wmma_f32_16x16x4_f32(
          false, a, false, bf, (short)0, acc, false, false);
    }
    int colk = k0 + l15;
#pragma unroll
    for (int r = 0; r < 8; ++r) {
      int m = r + half * 8;
      attRow[(size_t)m * NN + colk] = acc[r];
      lmax8[r] = fmaxf(lmax8[r], acc[r]);
    }
  }
#pragma unroll
  for (int r = 0; r < 8; ++r)
    rowmax_p[r + half * 8][wave * 16 + l15] = lmax8[r];

  __threadfence();   // make phase-A global stores visible across the WGP
  __syncthreads();

  if (tid < 16) {
    float m = -3.0e38f;
    for (int j = 0; j < 128; ++j) m = fmaxf(m, rowmax_p[tid][j]);
    rowred[tid] = m;
    conf[(size_t)b * NN + q0 + tid] = m;   // conf = max cosine (pre-tau)
  }
  __syncthreads();

  // --- Phase B/C: softmax, each wave owns 2 complete rows ---
  for (int rr = 0; rr < 2; ++rr) {
    int row = wave * 2 + rr;
    float rmax = rowred[row];
    float* rp = attRow + (size_t)row * NN;
    float s = 0.f;
    for (int col = lane; col < NN; col += 32) {
      float p = __expf((rp[col] - rmax) * TAU_INV);
      rp[col] = p;
      s += p;
    }
#pragma unroll
    for (int off = 16; off > 0; off >>= 1) s += __shfl_xor(s, off, 32);
    float rinv = 1.0f / s;
    for (int col = lane; col < NN; col += 32) rp[col] *= rinv;
  }
}

// ---------------------------------------------------------------------------
// Kernel 6: out GEMM  out[b][c][q] = gamma * sum_k pv[c][k]*att[q][k] + value
// Block = 256 threads computes a [128 c x 64 q] tile; K=4096 in 64-chunks
// staged through LDS (pv row-major, att tile read transposed from LDS).
// Wave w owns c-tile w and all four 16-wide q sub-tiles (4x v8f accum).
// ---------------------------------------------------------------------------
__global__ __launch_bounds__(256) void out_kernel(
    const float* __restrict__ pv,     // [b][c][k]
    const float* __restrict__ att,    // [b][q][k] (softmaxed)
    const float* __restrict__ value,  // [b][c][n]
    const float* __restrict__ gamma,
    float* __restrict__ outy) {       // [b][c][q]
  int b = blockIdx.y;
  int q1 = blockIdx.x * 64;
  int tid = threadIdx.x, wave = tid >> 5, lane = tid & 31;
  int half = lane >> 4, l15 = lane & 15;
  int c0 = wave * 16;

  __shared__ float pv_lds[128][66];
  __shared__ float at_lds[64][66];

  const float* pvb = pv + (size_t)b * NC * NN;
  const float* atb = att + ((size_t)b * NN + q1) * NN;

  v8f acc[4];
#pragma unroll
  for (int j = 0; j < 4; ++j) acc[j] = {};

  for (int kc = 0; kc < NN; kc += 64) {
    __syncthreads();
    // stage pv [128 x 64] (k-contiguous, coalesced)
    for (int i = tid; i < 128 * 64; i += 256) {
      int c = i >> 6, k = i & 63;
      pv_lds[c][k] = pvb[(size_t)c * NN + kc + k];
    }
    // stage att [64 q x 64 k] (k-contiguous, coalesced)
    for (int i = tid; i < 64 * 64; i += 256) {
      int q = i >> 6, k = i & 63;
      at_lds[q][k] = atb[(size_t)q * NN + kc + k];
    }
    __syncthreads();

    for (int kk = 0; kk < 64; kk += 4) {
      int ck = kk + half * 2;
      v2f a;
      a.x = pv_lds[c0 + l15][ck];
      a.y = pv_lds[c0 + l15][ck + 1];
#pragma unroll
      for (int j = 0; j < 4; ++j) {
        v2f bf;
        bf.x = at_lds[j * 16 + l15][ck];      // B[k][q] = att[q][k]
        bf.y = at_lds[j * 16 + l15][ck + 1];
        acc[j] = __builtin_amdgcn_wmma_f32_16x16x4_f32(
            false, a, false, bf, (short)0, acc[j], false, false);
      }
    }
  }

  float g = gamma[0];
#pragma unroll
  for (int j = 0; j < 4; ++j) {
#pragma unroll
    for (int r = 0; r < 8; ++r) {
      int m = c0 + r + half * 8;
      int q = q1 + j * 16 + l15;
      size_t idx = ((size_t)b * NC + m) * NN + q;
      outy[idx] = fmaf(g, acc[j][r], value[idx]);
    }
  }
}

// ---------------------------------------------------------------------------
// Host launcher.  Workspace layout (floats):
//   pq  [B][N][C]  8MB   (raw then normalized query proj)
//   pkn [B][C][N]  8MB   (raw then normalized key proj)
//   pv  [B][C][N]  8MB
//   mq  [B][C], mk [B][C]
// d_out layout: att (B*N*N) | conf (B*N) | out (B*C*N)
// ---------------------------------------------------------------------------
extern "C" void kernel_launch(void* const* d_in, const int* in_sizes, int n_in,
                              void* d_out, int out_size, void* d_ws, size_t ws_size,
                              hipStream_t stream) {
  const float* key   = (const float*)d_in[0];
  const float* query = (const float*)d_in[1];
  const float* value = (const float*)d_in[2];
  const float* qw = (const float*)d_in[3];
  const float* qb = (const float*)d_in[4];
  const float* kw = (const float*)d_in[5];
  const float* kb = (const float*)d_in[6];
  const float* vw = (const float*)d_in[7];
  const float* vb = (const float*)d_in[8];
  const float* gamma = (const float*)d_in[9];

  float* att  = (float*)d_out;
  float* conf = att + (size_t)NB * NN * NN;
  float* outy = conf + (size_t)NB * NN;

  float* ws  = (float*)d_ws;
  float* pq  = ws;
  float* pkn = pq + (size_t)NB * NN * NC;
  float* pv  = pkn + (size_t)NB * NC * NN;
  float* mq  = pv + (size_t)NB * NC * NN;
  float* mk  = mq + NB * NC;

  dim3 pblk(32, 8);
  dim3 pgrd(NN / 32, NC / 8, NB);
  proj_kernel<<<pgrd, pblk, 0, stream>>>(query, qw, qb, pq, 1);
  proj_kernel<<<pgrd, pblk, 0, stream>>>(key, kw, kb, pkn, 0);
  proj_kernel<<<pgrd, pblk, 0, stream>>>(value, vw, vb, pv, 0);

  mean_kernel<<<dim3(NC, NB), 256, 0, stream>>>(pq, pkn, mq, mk);
  norm_q_kernel<<<dim3(NN / 8, NB), 256, 0, stream>>>(pq, mq);
  norm_k_kernel<<<dim3(NN / 256, NB), 256, 0, stream>>>(pkn, mk);

  attn_kernel<<<dim3(NN / 16, NB), 256, 0, stream>>>(pq, pkn, att, conf);
  out_kernel<<<dim3(NN / 64, NB), 256, 0, stream>>>(pv, att, value, gamma, outy);
}